// PerHeadConvNetLayer_13125420056913
// MI455X (gfx1250) — compile-verified
//
#include <hip/hip_runtime.h>
#include <hip/hip_bf16.h>

typedef __attribute__((ext_vector_type(2))) float v2f;
typedef __attribute__((ext_vector_type(8))) float v8f;

// Native f32 WMMA on CDNA5: D(16x16,f32) = A(16x4,f32) x B(4x16,f32) + C
__device__ __forceinline__ v8f wmma_f32x4(v2f a, v2f b, v8f c) {
    return __builtin_amdgcn_wmma_f32_16x16x4_f32(false, a, false, b, (short)0, c,
                                                 false, false);
}

// Fast silu: v_exp_f32 + v_rcp_f32 instead of the IEEE div-scale chain.
__device__ __forceinline__ float silu(float v) {
    return v * __builtin_amdgcn_rcpf(1.0f + __expf(-v));
}

// ---------------------------------------------------------------------------
// Kernel 0: pre-scale weights once (amortized over ~40k wave-tiles that would
// otherwise v_pk_mul every B fragment).
//   W1s  = W1  / (sqrt(32)*sqrt(16)),  Wm1s = Wm1 / sqrt(8),  Wm2s = Wm2 / 8
// ---------------------------------------------------------------------------
__global__ void __launch_bounds__(256)
prep_weights_kernel(const float* __restrict__ W1,   // (3,32,32)  = 3072
                    const float* __restrict__ Wm1,  // (8,64)     = 512
                    const float* __restrict__ Wm2,  // (64,96)    = 6144
                    float* __restrict__ W1s,
                    float* __restrict__ Wm1s,
                    float* __restrict__ Wm2s)
{
    int i = blockIdx.x * blockDim.x + threadIdx.x;
    if (i < 3072) W1s[i]  = W1[i]  * 0.04419417382415922f;
    if (i < 512)  Wm1s[i] = Wm1[i] * 0.3535533905932738f;
    if (i < 6144) Wm2s[i] = Wm2[i] * 0.125f;
}

// ---------------------------------------------------------------------------
// Kernel 1: per-l node channel mixing (templated on compile-time geometry).
//   y[n, OFFM+m, v] = sum_u x[n, OFFX + u*D + m] * W1s[L][u,v]
// GEMM rows = N*D (n-major, m fastest), K = 32, 32 output cols.
// One wave: 16x32 tile = two 16x16 f32 WMMA accumulators, 8 K-steps of K=4.
// Loads are clamped (never predicated) so EXEC stays all-1s through the WMMAs.
// ---------------------------------------------------------------------------
template <int L, int D, int OFFX, int OFFM>
__global__ void __launch_bounds__(256)
node_transform_kernel(const float* __restrict__ x,    // (N,288)
                      const float* __restrict__ W1s,  // (3,32,32) pre-scaled
                      float* __restrict__ y,          // (N,9,32)
                      int rows)
{
    int lane = threadIdx.x & 31;
    int wv   = threadIdx.x >> 5;
    int tile = blockIdx.x * (blockDim.x >> 5) + wv;
    int base = tile * 16;
    if (base >= rows) return;                 // wave-uniform exit

    int lrow = lane & 15;                     // M (for A) / N-col (for B,D)
    int hi   = lane >> 4;                     // half-wave id
    int koff = hi << 1;                       // per-half K offset

    const float* Wl = W1s + L * 1024;         // W1s[L][u][v], u-major

    // A-row pointer for this lane; clamp keeps the address valid (no EXEC split)
    int r  = base + lrow;
    r      = r < rows ? r : rows - 1;
    int n  = r / D;                           // constant divisor -> mul/shift
    int ml = r % D;
    const float* arow = x + (size_t)n * 288 + OFFX + ml; // elem u at arow[u*D]

    v8f c0 = {};
    v8f c1 = {};
    #pragma unroll
    for (int k0 = 0; k0 < 32; k0 += 4) {
        int kb = k0 + koff;
        v2f a;
        a.x = arow[kb * D];                   // compile-time stride
        a.y = arow[(kb + 1) * D];
        v2f b0, b1;
        b0.x = Wl[kb * 32 + lrow];
        b0.y = Wl[(kb + 1) * 32 + lrow];
        b1.x = Wl[kb * 32 + 16 + lrow];
        b1.y = Wl[(kb + 1) * 32 + 16 + lrow];
        c0 = wmma_f32x4(a, b0, c0);
        c1 = wmma_f32x4(a, b1, c1);
    }

    // D layout: lane holds col v=lrow (c0) / 16+lrow (c1); VGPR i -> row i+hi*8
    #pragma unroll
    for (int i = 0; i < 8; ++i) {
        int rr = base + i + hi * 8;
        if (rr < rows) {
            int nn = rr / D, mm = rr % D;
            float* yr = y + (size_t)nn * 288 + (size_t)(OFFM + mm) * 32;
            yr[lrow]      = c0[i];
            yr[16 + lrow] = c1[i];
        }
    }
}

// ---------------------------------------------------------------------------
// Kernel 2: fused edge MLP (WMMA) + tensor-product message + atomic scatter.
// One wave owns 16 edges. 8 waves / block. LDS: per-wave 16x96 f32 tile.
// ---------------------------------------------------------------------------
__global__ void __launch_bounds__(256)
edge_kernel(const float* __restrict__ emb,    // (E,8)
            const float* __restrict__ eattr,  // (E,9)
            const int*   __restrict__ esrc,   // (E,)
            const int*   __restrict__ edst,   // (E,)
            const float* __restrict__ Wm1s,   // (8,64)  pre-scaled
            const float* __restrict__ Wm2s,   // (64,96) pre-scaled
            const float* __restrict__ y,      // (N,9,32)
            float*       __restrict__ acc,    // (N,96) zero-initialized
            int E)
{
    __shared__ float lds[8][16 * 96];         // 48 KB
    int lane = threadIdx.x & 31;
    int wv   = threadIdx.x >> 5;
    int tile = blockIdx.x * 8 + wv;
    int e0   = tile * 16;
    if (e0 >= E) return;                      // wave-uniform exit

    int lrow = lane & 15;
    int hi   = lane >> 4;
    int koff = hi << 1;
    float* myl = lds[wv];

    // ---- layer 1: h = silu((16x8) @ (8x64)) ----
    int er = e0 + lrow;
    er = er < E ? er : E - 1;                 // clamp: no EXEC divergence
    const float* arow = emb + (size_t)er * 8;
    v8f h[4] = {};
    #pragma unroll
    for (int k0 = 0; k0 < 8; k0 += 4) {
        int kb = k0 + koff;
        v2f a;
        a.x = arow[kb];
        a.y = arow[kb + 1];
        #pragma unroll
        for (int j = 0; j < 4; ++j) {
            v2f b;
            b.x = Wm1s[kb * 64 + j * 16 + lrow];
            b.y = Wm1s[(kb + 1) * 64 + j * 16 + lrow];
            h[j] = wmma_f32x4(a, b, h[j]);
        }
    }
    // silu + transpose into LDS as h[M][k] (row-major, 16x64)
    #pragma unroll
    for (int j = 0; j < 4; ++j)
        #pragma unroll
        for (int i = 0; i < 8; ++i)
            myl[(i + hi * 8) * 64 + j * 16 + lrow] = silu(h[j][i]);

    // ---- layer 2: ew = (16x64) @ (64x96) ----
    v8f ew[6] = {};
    #pragma unroll 4
    for (int k0 = 0; k0 < 64; k0 += 4) {
        int kb = k0 + koff;
        v2f a;                                 // same-wave LDS RAW: in-order DS
        a.x = myl[lrow * 64 + kb];
        a.y = myl[lrow * 64 + kb + 1];
        #pragma unroll
        for (int j = 0; j < 6; ++j) {
            v2f b;
            b.x = Wm2s[kb * 96 + j * 16 + lrow];
            b.y = Wm2s[(kb + 1) * 96 + j * 16 + lrow];
            ew[j] = wmma_f32x4(a, b, ew[j]);
        }
    }
    // transpose ew into LDS as ewT[e_local][k] (16x96); overwrites h region
    #pragma unroll
    for (int j = 0; j < 6; ++j)
        #pragma unroll
        for (int i = 0; i < 8; ++i)
            myl[(i + hi * 8) * 96 + j * 16 + lrow] = ew[j][i];

    asm volatile("s_wait_dscnt 0" ::: "memory");   // all DS stores visible

    // ---- message + scatter: lane = channel u ----
    const float cg1 = 0.5773502691896258f;    // 1/sqrt(3)
    const float cg2 = 0.4472135954999579f;    // 1/sqrt(5)
    for (int ee = 0; ee < 16; ++ee) {
        int e = e0 + ee;
        if (e >= E) break;
        int src = esrc[e];
        int dst = edst[e];
        if (ee + 1 < 16 && e + 1 < E) {       // L2 prefetch of next source row
            int srcn = esrc[e + 1];
            __builtin_prefetch(y + (size_t)srcn * 288, 0, 0);
        }
        const float* ea = eattr + (size_t)e * 9;  // uniform -> broadcast loads
        const float* ys = y + (size_t)src * 288 + lane;
        float d0 = ys[0] * ea[0];
        float d1 = ys[32] * ea[1] + ys[64] * ea[2] + ys[96] * ea[3];
        float d2 = ys[128] * ea[4] + ys[160] * ea[5] + ys[192] * ea[6]
                 + ys[224] * ea[7] + ys[256] * ea[8];
        float m0 = myl[ee * 96 + lane]      * d0;
        float m1 = myl[ee * 96 + 32 + lane] * d1 * cg1;
        float m2 = myl[ee * 96 + 64 + lane] * d2 * cg2;
        float* ap = acc + (size_t)dst * 96 + lane;
        atomicAdd(ap,      m0);               // global_atomic_add_f32, no return
        atomicAdd(ap + 32, m1);
        atomicAdd(ap + 64, m2);
    }
}

// ---------------------------------------------------------------------------
// Kernel 3: per-node heads: silu(acc @ W2 * norm + sc), sc folded in (one-hot).
// One wave per node; lane = output channel k.
// ---------------------------------------------------------------------------
__global__ void __launch_bounds__(256)
head_kernel(const float* __restrict__ x,      // (N,288) (x0 = x[:, :32])
            const float* __restrict__ attrs,  // (N,4)
            const float* __restrict__ W2h0,   // (64,32)
            const float* __restrict__ W2h1,   // (96,32)
            const float* __restrict__ Wsc0,   // (32,4,32)
            const float* __restrict__ Wsc1,   // (32,4,32)
            const float* __restrict__ acc,    // (N,96)
            float*       __restrict__ out,    // (2,N,32)
            int N)
{
    int lane = threadIdx.x & 31;
    int wv   = threadIdx.x >> 5;
    int n = blockIdx.x * (blockDim.x >> 5) + wv;
    if (n >= N) return;

    const float scn = 0.08838834764831845f;   // 1/sqrt(32*4)
    const float n0  = 0.125f;                 // 1/sqrt(64)
    const float n1  = 0.10206207261596575f;   // 1/sqrt(96)

    float a0 = attrs[(size_t)n * 4 + 0];
    float a1 = attrs[(size_t)n * 4 + 1];
    float a2 = attrs[(size_t)n * 4 + 2];
    float a3 = attrs[(size_t)n * 4 + 3];

    float sc0 = 0.0f, sc1 = 0.0f;
    const float* x0 = x + (size_t)n * 288;    // block l=0, d=1: x0[u] = x[n,u]
    #pragma unroll 4
    for (int u = 0; u < 32; ++u) {
        float xv = x0[u];                     // uniform -> broadcast
        const float* w0 = Wsc0 + u * 128;
        const float* w1 = Wsc1 + u * 128;
        sc0 += xv * (a0 * w0[lane] + a1 * w0[32 + lane] +
                     a2 * w0[64 + lane] + a3 * w0[96 + lane]);
        sc1 += xv * (a0 * w1[lane] + a1 * w1[32 + lane] +
                     a2 * w1[64 + lane] + a3 * w1[96 + lane]);
    }
    sc0 *= scn; sc1 *= scn;

    const float* an = acc + (size_t)n * 96;
    float o0 = 0.0f, o1 = 0.0f;
    #pragma unroll 8
    for (int j = 0; j < 64; ++j) {
        float av = an[j];                     // uniform -> broadcast
        o0 += av * W2h0[j * 32 + lane];
        o1 += av * W2h1[j * 32 + lane];
    }
    #pragma unroll 8
    for (int j = 64; j < 96; ++j)
        o1 += an[j] * W2h1[j * 32 + lane];

    o0 = silu(o0 * n0 + sc0);
    o1 = silu(o1 * n1 + sc1);
    out[(size_t)n * 32 + lane]                   = o0;
    out[(size_t)N * 32 + (size_t)n * 32 + lane]  = o1;
}

// ---------------------------------------------------------------------------
extern "C" void kernel_launch(void* const* d_in, const int* in_sizes, int n_in,
                              void* d_out, int out_size, void* d_ws, size_t ws_size,
                              hipStream_t stream)
{
    const float* x     = (const float*)d_in[0];
    const float* attrs = (const float*)d_in[1];
    const float* eattr = (const float*)d_in[2];
    const float* emb   = (const float*)d_in[3];
    const int*   esrc  = (const int*)d_in[4];
    const int*   edst  = (const int*)d_in[5];
    const float* W1    = (const float*)d_in[6];
    const float* Wm1   = (const float*)d_in[7];
    const float* Wm2   = (const float*)d_in[8];
    const float* W2h0  = (const float*)d_in[9];
    const float* W2h1  = (const float*)d_in[10];
    const float* Wsc0  = (const float*)d_in[11];
    const float* Wsc1  = (const float*)d_in[12];

    int N = in_sizes[0] / 288;
    int E = in_sizes[4];

    float* y    = (float*)d_ws;                  // N*288 floats
    float* acc  = y   + (size_t)N * 288;         // N*96 floats
    float* W1s  = acc + (size_t)N * 96;          // 3072 floats
    float* Wm1s = W1s + 3072;                    // 512 floats
    float* Wm2s = Wm1s + 512;                    // 6144 floats

    hipMemsetAsync(acc, 0, (size_t)N * 96 * sizeof(float), stream);

    prep_weights_kernel<<<(6144 + 255) / 256, 256, 0, stream>>>(
        W1, Wm1, Wm2, W1s, Wm1s, Wm2s);

    // Node transform: one templated launch per l
    {
        int rows   = N * 1;
        int blocks = ((rows + 15) / 16 + 7) / 8;
        node_transform_kernel<0, 1, 0, 0><<<blocks, 256, 0, stream>>>(x, W1s, y, rows);
    }
    {
        int rows   = N * 3;
        int blocks = ((rows + 15) / 16 + 7) / 8;
        node_transform_kernel<1, 3, 32, 1><<<blocks, 256, 0, stream>>>(x, W1s, y, rows);
    }
    {
        int rows   = N * 5;
        int blocks = ((rows + 15) / 16 + 7) / 8;
        node_transform_kernel<2, 5, 128, 4><<<blocks, 256, 0, stream>>>(x, W1s, y, rows);
    }

    int etiles  = (E + 15) / 16;
    int eblocks = (etiles + 7) / 8;
    edge_kernel<<<eblocks, 256, 0, stream>>>(emb, eattr, esrc, edst,
                                             Wm1s, Wm2s, y, acc, E);

    int nblocks = (N + 7) / 8;
    head_kernel<<<nblocks, 256, 0, stream>>>(x, attrs, W2h0, W2h1,
                                             Wsc0, Wsc1, acc,
                                             (float*)d_out, N);
}